// EnergyXL_63273458204875
// MI455X (gfx1250) — compile-verified
//
#include <hip/hip_runtime.h>
#include <hip/hip_bf16.h>

// Problem constants (from reference)
#define B_MOL   256
#define A_ATOMS 100
#define NDIM    400                    // 4*A
#define NELEM   (NDIM * NDIM)          // 160000
#define NPAIR   4950                   // A*(A-1)/2
#define GSS_C   12.23f
#define HSP_C   2.65f

// Each task = one 4-wide sub-row of a 4x4 atom block: t = i*100 + ja,
// covering matrix elements (i, 4*ja .. 4*ja+3). Element offset == 4*t.
#define NTASK   (NDIM * A_ATOMS)       // 40000 tasks per molecule
#define SPLIT   4                      // workgroups per molecule
#define TPB     512                    // 16 waves (wave32)
#define TCHUNK  (NTASK / SPLIT)        // 10000 tasks per workgroup

typedef float v2f __attribute__((ext_vector_type(2)));
typedef float v4f __attribute__((ext_vector_type(4)));
typedef float v8f __attribute__((ext_vector_type(8)));

// One workgroup per (chunk, molecule). Streams the block-upper-triangle of
// P/D once via b128 loads, M_pair once (lower-triangle contributions folded
// via symmetry of P), reduces to a single partial per workgroup.
__global__ __launch_bounds__(TPB)
void EnergyXL_fock_kernel(const float* __restrict__ Mdiag,
                          const float* __restrict__ Mpair,
                          const float* __restrict__ P,
                          const float* __restrict__ D,
                          float* __restrict__ ws_partial) {
  const int b     = blockIdx.y;
  const int chunk = blockIdx.x;
  const int tid   = threadIdx.x;

  const v4f* __restrict__ Pb4 = (const v4f*)(P + (size_t)b * NELEM);
  const v4f* __restrict__ Db4 = (const v4f*)(D + (size_t)b * NELEM);
  const v4f* __restrict__ Mp4 = (const v4f*)(Mpair + (size_t)b * (NPAIR * 16));
  const v4f* __restrict__ Md4 = (const v4f*)(Mdiag + (size_t)b * (A_ATOMS * 16));

  float s = 0.f;
  const int t_end = (chunk + 1) * TCHUNK;
  for (int t = chunk * TCHUNK + tid; t < t_end; t += TPB) {
    const int i  = t / A_ATOMS;        // matrix row (0..399)
    const int ja = t - i * A_ATOMS;    // atom block column (0..99)
    const int ia = i >> 2;             // atom block row
    const int r  = i & 3;              // sub-row within block

    if (ja > ia) {
      // Upper pair block; folds mirrored lower-triangle term (P symmetric):
      //   contrib = d*F + 0.25*HSP*p^2 + 0.5*Hc*p,  F = Hc - 0.5*HSP*p
      const int pidx = ia * (A_ATOMS - 1) - ((ia * (ia - 1)) >> 1) + (ja - ia - 1);
      __builtin_prefetch(&Pb4[t + TPB], 0, 3);   // global_prefetch_b8 (linear)
      __builtin_prefetch(&Db4[t + TPB], 0, 3);
      const v4f m4 = Mp4[pidx * 4 + r];          // M_pair block row r (b128)
      const v4f p4 = Pb4[t];                     // P  sub-row (b128)
      const v4f d4 = Db4[t];                     // D  sub-row (b128)
#pragma unroll
      for (int c = 0; c < 4; ++c) {
        const float F = fmaf(-0.5f * HSP_C, p4[c], m4[c]);
        s += d4[c] * F + (0.25f * HSP_C) * p4[c] * p4[c] + 0.5f * m4[c] * p4[c];
      }
    } else if (ja == ia) {
      // Diagonal atom block: load full 4x4 block (rows give Hc and transpose)
      const v4f p4 = Pb4[t];
      const v4f d4 = Db4[t];
      const v4f row0 = Md4[ia * 4 + 0];
      const v4f row1 = Md4[ia * 4 + 1];
      const v4f row2 = Md4[ia * 4 + 2];
      const v4f row3 = Md4[ia * 4 + 3];
      const v4f rows[4] = { row0, row1, row2, row3 };
      const v4f hr = rows[r];                    // Hc row: Md[ia][r][*]
#pragma unroll
      for (int c = 0; c < 4; ++c) {
        const float Hc = hr[c];
        const float p  = p4[c];
        const float F  = fmaf(GSS_C, p, Hc);
        float contrib  = d4[c] * F - (0.5f * GSS_C) * p * p;
        if (r > c)                               // h = Md[ia][c][r] below diag
          contrib -= 0.5f * (Hc - rows[c][r]) * p;
        s += contrib;
      }
    }
    // ja < ia: contribution already folded into the mirrored upper element.
  }

  // ---- reduction: 32 lanes -> 1 per wave (wave32 shuffles) ----
  for (int off = 16; off > 0; off >>= 1)
    s += __shfl_xor(s, off, 32);

  __shared__ float wsum[32];
  const int lane = tid & 31;
  const int wave = tid >> 5;
  if (tid < 32) wsum[tid] = 0.f;
  __syncthreads();
  if (lane == 0) wsum[wave] = s;                 // TPB/32 = 16 wave partials
  __syncthreads();

  // ---- final 16-partial reduction via V_WMMA_F32_16X16X4_F32 ----
  // A (16x4 f32): A[m,0] = wsum[m] (lanes 0-15 of VGPR0), rest 0.
  // B (4x16): all ones (layout-independent). D[m,n] = wsum[m], exact f32.
  if (wave == 0) {
    const float a0 = wsum[lane];                 // lanes 16-31 feed K=2 -> 0
    v2f Amat = { (lane < 16) ? a0 : 0.f, 0.f };
    v2f Bmat = { 1.f, 1.f };
    v8f Cmat = {};
    Cmat = __builtin_amdgcn_wmma_f32_16x16x4_f32(
        /*neg_a=*/false, Amat, /*neg_b=*/false, Bmat,
        /*c_mod=*/(short)0, Cmat, /*reuse_a=*/false, /*reuse_b=*/false);
    // Lane L<16 holds D[0..7, L] in C[0..7]; lane L+16 holds D[8..15, L].
    float tsum = ((Cmat[0] + Cmat[1]) + (Cmat[2] + Cmat[3])) +
                 ((Cmat[4] + Cmat[5]) + (Cmat[6] + Cmat[7]));
    tsum += __shfl_xor(tsum, 16, 32);            // fold D[8..15] half
    if (lane == 0) ws_partial[b * SPLIT + chunk] = tsum;
  }
}

// Sum the SPLIT partials per molecule -> Eelec[b]
__global__ void EnergyXL_finalize_kernel(const float* __restrict__ ws,
                                         float* __restrict__ out) {
  const int t = threadIdx.x + blockIdx.x * blockDim.x;
  if (t < B_MOL) {
    out[t] = (ws[t * SPLIT + 0] + ws[t * SPLIT + 1]) +
             (ws[t * SPLIT + 2] + ws[t * SPLIT + 3]);
  }
}

extern "C" void kernel_launch(void* const* d_in, const int* in_sizes, int n_in,
                              void* d_out, int out_size, void* d_ws, size_t ws_size,
                              hipStream_t stream) {
  (void)in_sizes; (void)n_in; (void)out_size; (void)ws_size;
  const float* Mdiag = (const float*)d_in[0];   // [B*A,4,4]
  const float* Mpair = (const float*)d_in[1];   // [B*4950,4,4]
  const float* P     = (const float*)d_in[2];   // [B,400,400] (symmetric)
  const float* D     = (const float*)d_in[3];   // [B,400,400] (symmetric)
  // d_in[4] = maskd, d_in[5] = mask: index structure is deterministic and
  // reproduced analytically in-kernel; not needed.
  float* out = (float*)d_out;                   // [B]
  float* ws  = (float*)d_ws;                    // SPLIT*B floats of scratch

  dim3 grid(SPLIT, B_MOL);
  EnergyXL_fock_kernel<<<grid, TPB, 0, stream>>>(Mdiag, Mpair, P, D, ws);
  EnergyXL_finalize_kernel<<<1, 256, 0, stream>>>(ws, out);
}